// FilteredBackProjection_58299886076393
// MI455X (gfx1250) — compile-verified
//
#include <hip/hip_runtime.h>
#include <hip/hip_bf16.h>
#include <math.h>

// ---------------------------------------------------------------------------
// FBP: sinogram (2,1,360,363) fp32 -> recon (2,1,256,256) fp32.
//
// Stage 0: zero-pad sinogram rows to K=364 (branch-free WMMA K-loop).
// Stage 1: h[n] = real(ifft(2*|fftfreq(D)|)); scaled cos/sin tables.
// Stage 2: filtered = sino @ Circulant(h) via V_WMMA_F32_16X16X4_F32,
//          REVERSED doubled-h LDS table -> B pair at increasing adjacent
//          addresses, so ds_load_2addr lands in operand order (no swaps).
// Stage 3: backprojection; zero-padded LDS rows remove all bounds checks.
// ---------------------------------------------------------------------------

#define IMG 256
#define NA 360
#define DD 363          // detectors
#define KPAD 364        // padded K (multiple of 4)
#define FSTRIDE 368     // 23*16, padded row stride of filtered sinogram
#define NROWS 720       // B*C*A
#define MTILES 45       // 720/16
#define NTILES 23       // ceil(363/16)
#define NTILES_TOT (MTILES * NTILES)
#define CHUNK 16        // angles per LDS chunk in backprojection
#define PAD 80          // detector pad so u in [-75.94,437.94] never branches
#define ROWL 520        // PAD + 438 + 1 = 519 -> 520

typedef float v2f __attribute__((ext_vector_type(2)));
typedef float v8f __attribute__((ext_vector_type(8)));

// ---------------- Stage 0: pad sinogram K 363 -> 364 with zeros ------------
__global__ void fbp_pad_kernel(const float* __restrict__ sino,
                               float* __restrict__ sinop) {
    int idx = blockIdx.x * 256 + threadIdx.x;
    if (idx >= NROWS * KPAD) return;
    int row = idx / KPAD;
    int k   = idx - row * KPAD;
    sinop[idx] = (k < DD) ? sino[row * DD + k] : 0.0f;
}

// ---------------- Stage 1: ramp kernel + scaled trig tables ----------------
__global__ void fbp_prep_kernel(float* __restrict__ h, float* __restrict__ trig) {
    const int t = threadIdx.x;
    if (t < DD) {
        // h[n] = (1/D) * sum_k filt[k]*cos(2*pi*k*n/D), filt[k] = 2*min(k,D-k)/D
        const float c = 6.283185307179586f / (float)DD;
        float acc = 0.0f;
        for (int k = 0; k < DD; ++k) {
            int m  = (k * t) % DD;                 // keep cos arg in [0, 2*pi)
            int kk = (k < DD - k) ? k : (DD - k);
            acc += (2.0f * (float)kk / (float)DD) * cosf(c * (float)m);
        }
        h[t] = acc / (float)DD;
    }
    if (t < NA) {
        // u = x*cos(a)*q + y*sin(a)*q + 181,  q = 181 / scale
        const float scale = (127.5f * 1.4142135623730951f) / 181.0f;
        const float q = 181.0f / scale;
        float ang = 3.14159265358979f * (float)t / (float)NA;
        trig[t]      = cosf(ang) * q;
        trig[t + NA] = sinf(ang) * q;
    }
}

// ---------------- Stage 2: circulant GEMM via V_WMMA_F32_16X16X4_F32 -------
// filtered[row][d] = sum_k sinop[row][k] * h[(d-k) mod D]
// One wave32 per 16x16 tile of the 720x368 output. Branch-free, pipelined.
__global__ void __launch_bounds__(256)
fbp_filter_kernel(const float* __restrict__ sinop,
                  const float* __restrict__ h,
                  float* __restrict__ filtered) {
    // reversed doubled circulant table: hr[i] = h[(730 - i) mod 363]
    __shared__ float hr[736];
    for (int i = threadIdx.x; i < 736; i += 256) {
        int m = 730 - i;                    // [-5, 730]
        if (m < 0)   m += DD;
        if (m >= DD) m -= DD;
        if (m >= DD) m -= DD;
        hr[i] = h[m];
    }
    __syncthreads();

    const int wave = blockIdx.x * 8 + (threadIdx.x >> 5);
    if (wave >= NTILES_TOT) return;         // wave-uniform; EXEC stays all-1s

    const int lane  = threadIdx.x & 31;
    const int half  = lane >> 4;            // 0: K pair {0,1}, 1: K pair {2,3}
    const int l15   = lane & 15;
    const int tileM = wave / NTILES;
    const int tileN = wave % NTILES;

    // A: M = lane&15, K = kb + 2*half + {0,1}; 8B-aligned b64 loads.
    const float* arow = sinop + (tileM * 16 + l15) * KPAD + 2 * half;
    // B: N = lane&15; hr index increases with kb, pair is adjacent ascending:
    //   b.x = h[(dcol-k0) mod D] = hr[rbase+kb], b.y = hr[rbase+kb+1]
    const int dcol = tileN * 16 + l15;
    const float* bb = hr + (367 - dcol + 2 * half);   // rbase in [0, 369]

    v8f acc = {};
    v2f a  = *(const v2f*)arow;
    float bx = bb[0];
    float by = bb[1];

    for (int kb = 0; kb < KPAD - 4; kb += 4) {
        // prefetch next iteration's operands, then issue WMMA on current
        v2f   an  = *(const v2f*)(arow + kb + 4);
        float bxn = bb[kb + 4];
        float byn = bb[kb + 5];
        v2f b; b.x = bx; b.y = by;
        acc = __builtin_amdgcn_wmma_f32_16x16x4_f32(
            false, a, false, b, (short)0, acc, false, false);
        a = an; bx = bxn; by = byn;
    }
    {
        v2f b; b.x = bx; b.y = by;
        acc = __builtin_amdgcn_wmma_f32_16x16x4_f32(
            false, a, false, b, (short)0, acc, false, false);
    }

    // C/D layout: VGPR r holds M = r + 8*half, N = lane&15
    #pragma unroll
    for (int r = 0; r < 8; ++r) {
        filtered[(tileM * 16 + r + 8 * half) * FSTRIDE + dcol] = acc[r];
    }
}

// ---------------- Stage 3: backprojection, branch-free LDS gathers ---------
__global__ void __launch_bounds__(256)
fbp_backproj_kernel(const float* __restrict__ filtered,
                    const float* __restrict__ trig,
                    float* __restrict__ out) {
    __shared__ float srow[2][CHUNK][ROWL];  // 66560 B, zero-padded rows
    __shared__ float ctab[NA];
    __shared__ float stab[NA];

    const int tid = threadIdx.x;
    for (int i = tid; i < NA; i += 256) {
        ctab[i] = trig[i];
        stab[i] = trig[i + NA];
    }
    // zero the pads once; they are never overwritten by the chunk loads
    float* sflat = &srow[0][0][0];
    for (int i = tid; i < 2 * CHUNK * ROWL; i += 256) sflat[i] = 0.0f;

    const int bT = blockIdx.x;                  // 256 blocks of 16x16 pixels
    const int pi = (bT >> 4) * 16 + (tid >> 4); // image row -> y
    const int pj = (bT & 15) * 16 + (tid & 15); // image col -> x
    const float y = -1.0f + 2.0f * (float)pi / 255.0f;
    const float x = -1.0f + 2.0f * (float)pj / 255.0f;

    float acc0 = 0.0f, acc1 = 0.0f;

    for (int a0 = 0; a0 < NA; a0 += CHUNK) {
        __syncthreads();   // previous chunk consumed / pads zeroed
        #pragma unroll
        for (int r = 0; r < 2 * CHUNK; ++r) {
            const int b  = r >> 4;
            const int al = r & (CHUNK - 1);
            const float* src = filtered + (b * NA + a0 + al) * FSTRIDE;
            float* dst = &srow[b][al][PAD];
            for (int d = tid; d < DD; d += 256) dst[d] = src[d];
        }
        __syncthreads();

        #pragma unroll 4
        for (int al = 0; al < CHUNK; ++al) {
            const int a = a0 + al;
            // u shifted by PAD: up in [4.06, 517.94] -> no bounds checks
            const float up = x * ctab[a] + y * stab[a] + (181.0f + (float)PAD);
            const float fi = floorf(up);
            const float w1 = up - fi;
            const float w0 = 1.0f - w1;
            const int   i0 = (int)fi;
            acc0 = fmaf(srow[0][al][i0], w0, fmaf(srow[0][al][i0 + 1], w1, acc0));
            acc1 = fmaf(srow[1][al][i0], w0, fmaf(srow[1][al][i0 + 1], w1, acc1));
        }
    }

    const float sc = 3.14159265358979f / (float)NA;
    out[pi * IMG + pj]             = acc0 * sc;   // batch 0
    out[IMG * IMG + pi * IMG + pj] = acc1 * sc;   // batch 1
}

// ---------------------------------------------------------------------------
extern "C" void kernel_launch(void* const* d_in, const int* in_sizes, int n_in,
                              void* d_out, int out_size, void* d_ws, size_t ws_size,
                              hipStream_t stream) {
    (void)in_sizes; (void)n_in; (void)out_size; (void)ws_size;
    const float* sino = (const float*)d_in[0];   // (2,1,360,363) fp32
    float* ws       = (float*)d_ws;
    float* h        = ws;                 // 363 floats (pad to 512)
    float* trig     = ws + 512;           // 720 floats (pad to 1280)
    float* sinop    = ws + 1280;          // 720*364 = 262080 (pad to 262144)
    float* filtered = ws + 1280 + 262144; // 720*368 = 264960 floats

    fbp_prep_kernel<<<1, 512, 0, stream>>>(h, trig);
    fbp_pad_kernel<<<(NROWS * KPAD + 255) / 256, 256, 0, stream>>>(sino, sinop);
    fbp_filter_kernel<<<(NTILES_TOT + 7) / 8, 256, 0, stream>>>(sinop, h, filtered);
    fbp_backproj_kernel<<<IMG * IMG / 256, 256, 0, stream>>>(
        filtered, trig, (float*)d_out);
}